// LinearAttention_2130303779058
// MI455X (gfx1250) — compile-verified
//
#include <hip/hip_runtime.h>

typedef float v2f __attribute__((ext_vector_type(2)));
typedef float v8f __attribute__((ext_vector_type(8)));

#define NB    16     // batches
#define DK    64
#define NSEQ  8192
#define DV    64
#define SPLITS 16
#define ROWS_PER_SPLIT (NSEQ / SPLITS)  // 512
#define CHUNK 64
#define NCHUNKS (ROWS_PER_SPLIT / CHUNK) // 8

// Async DMA: global -> LDS, bypassing VGPRs. Tracked with ASYNCcnt.
__device__ __forceinline__ void async_load_b128(unsigned lds_off,
                                                const float* gaddr) {
  asm volatile("global_load_async_to_lds_b128 %0, %1, off"
               :: "v"(lds_off), "v"(gaddr) : "memory");
}
#define WAIT_ASYNC_LE(N) asm volatile("s_wait_asynccnt " #N ::: "memory")

__device__ __forceinline__ unsigned lds_off32(const void* p) {
  return (unsigned)(unsigned long long)p;  // low 32 bits of flat addr = LDS offset
}

// ---------------- Phase 1: partial K^T V per (batch, split) ----------------
// Each block: one (b, split). 16 waves, wave (mi,vi) owns 16x16 tile of [64x64].
// Double-buffered LDS filled by async-to-LDS DMA; WMMA consumes from LDS.
__global__ __launch_bounds__(512) void ktv_partial_kernel(
    const float* __restrict__ k, const float* __restrict__ v,
    float* __restrict__ partial) {
  __shared__ float ldsK[2][CHUNK * DK];  // [buf][n_local][kdim]
  __shared__ float ldsV[2][CHUNK * DV];  // [buf][n_local][vdim]

  const int blk   = blockIdx.x;
  const int b     = blk / SPLITS;
  const int split = blk % SPLITS;
  const int nbase = split * ROWS_PER_SPLIT;

  const int tid  = threadIdx.x;
  const int wave = tid >> 5;
  const int lane = tid & 31;
  const int half = lane >> 4;      // selects K pair {2h, 2h+1}
  const int l    = lane & 15;
  const int mi   = wave >> 2;      // kdim tile 0..3
  const int vi   = wave & 3;       // vdim tile 0..3

  const float* kb = k + (size_t)b * NSEQ * DK;
  const float* vb = v + (size_t)b * NSEQ * DV;

  // this thread's fixed slice of each chunk: two float4 rows for k and v
  const int r0f = tid >> 4;              // f = tid      -> row
  const int r1f = (tid + 512) >> 4;      // f = tid+512  -> row
  const int c4  = (tid & 15) << 2;       // float column

  // issue async fills for chunk 0 into buffer 0
  {
    const int n0 = nbase;
    async_load_b128(lds_off32(&ldsK[0][r0f * DK + c4]),
                    &kb[(size_t)(n0 + r0f) * DK + c4]);
    async_load_b128(lds_off32(&ldsK[0][r1f * DK + c4]),
                    &kb[(size_t)(n0 + r1f) * DK + c4]);
    async_load_b128(lds_off32(&ldsV[0][r0f * DV + c4]),
                    &vb[(size_t)(n0 + r0f) * DV + c4]);
    async_load_b128(lds_off32(&ldsV[0][r1f * DV + c4]),
                    &vb[(size_t)(n0 + r1f) * DV + c4]);
  }

  v8f acc = {};

  for (int c = 0; c < NCHUNKS; ++c) {
    const int cur = c & 1;
    if (c + 1 < NCHUNKS) {
      // issue DMA for next chunk into the other buffer (safe: last readers of
      // that buffer finished at the trailing barrier of iteration c-1)
      const int nxt = (c + 1) & 1;
      const int n1  = nbase + (c + 1) * CHUNK;
      async_load_b128(lds_off32(&ldsK[nxt][r0f * DK + c4]),
                      &kb[(size_t)(n1 + r0f) * DK + c4]);
      async_load_b128(lds_off32(&ldsK[nxt][r1f * DK + c4]),
                      &kb[(size_t)(n1 + r1f) * DK + c4]);
      async_load_b128(lds_off32(&ldsV[nxt][r0f * DV + c4]),
                      &vb[(size_t)(n1 + r0f) * DV + c4]);
      async_load_b128(lds_off32(&ldsV[nxt][r1f * DV + c4]),
                      &vb[(size_t)(n1 + r1f) * DV + c4]);
      WAIT_ASYNC_LE(0x4);   // 4 newest (next chunk) may remain outstanding
    } else {
      WAIT_ASYNC_LE(0x0);
    }
    __syncthreads();        // all waves' DMA for buffer `cur` landed

    // A = K^T tile [M=kdim 16][K=4 n-steps], B = V tile [K=4][N=vdim 16]
#pragma unroll
    for (int s = 0; s < CHUNK; s += 4) {
      const int r0 = s + half * 2;
      v2f a, bm;
      a.x  = ldsK[cur][(r0 + 0) * DK + mi * 16 + l];
      a.y  = ldsK[cur][(r0 + 1) * DK + mi * 16 + l];
      bm.x = ldsV[cur][(r0 + 0) * DV + vi * 16 + l];
      bm.y = ldsV[cur][(r0 + 1) * DV + vi * 16 + l];
      acc = __builtin_amdgcn_wmma_f32_16x16x4_f32(
          /*neg_a=*/false, a, /*neg_b=*/false, bm,
          /*c_mod=*/(short)0, acc, /*reuse_a=*/false, /*reuse_b=*/false);
    }
    __syncthreads();        // everyone done reading `cur` before it is refilled
  }

  // D layout: VGPR j -> row (mi*16 + j + 8*half), col (vi*16 + l)
  float* p = partial + (size_t)blk * (DK * DV);
#pragma unroll
  for (int j = 0; j < 8; ++j)
    p[(mi * 16 + j + 8 * half) * DV + vi * 16 + l] = acc[j];
}

// ---------------- Phase 1.5: deterministic reduction of split partials ------
__global__ __launch_bounds__(256) void ktv_reduce_kernel(
    const float* __restrict__ partial, float* __restrict__ ktv) {
  const int b = blockIdx.x;
  for (int e = threadIdx.x; e < DK * DV; e += 256) {
    float s = 0.f;
#pragma unroll
    for (int sp = 0; sp < SPLITS; ++sp)
      s += partial[(size_t)(b * SPLITS + sp) * (DK * DV) + e];
    ktv[(size_t)b * (DK * DV) + e] = s;
  }
}

// ---------------- Phase 2: out[b] = Q[b]^T * ktv[b] -------------------------
// Each block: (n-block of 64 rows, batch). q tile + ktv DMA'd into LDS.
__global__ __launch_bounds__(512) void qktv_kernel(
    const float* __restrict__ q, const float* __restrict__ ktv,
    float* __restrict__ out) {
  __shared__ float ldsQ[DK * 64];   // [kdim][n_local]
  __shared__ float ldsT[DK * DV];   // [kdim][vdim]

  const int nbase = blockIdx.x * 64;
  const int b     = blockIdx.y;

  const int tid  = threadIdx.x;
  const int wave = tid >> 5;
  const int lane = tid & 31;
  const int half = lane >> 4;
  const int l    = lane & 15;
  const int ni   = wave >> 2;      // n tile 0..3
  const int vi   = wave & 3;       // v tile 0..3

  const float* qb = q + (size_t)b * DK * NSEQ;
  const float* tb = ktv + (size_t)b * DK * DV;

  {
    const int r0f = tid >> 4;
    const int r1f = (tid + 512) >> 4;
    const int c4  = (tid & 15) << 2;
    async_load_b128(lds_off32(&ldsQ[r0f * 64 + c4]),
                    &qb[(size_t)r0f * NSEQ + nbase + c4]);
    async_load_b128(lds_off32(&ldsQ[r1f * 64 + c4]),
                    &qb[(size_t)r1f * NSEQ + nbase + c4]);
    async_load_b128(lds_off32(&ldsT[r0f * DV + c4]), &tb[r0f * DV + c4]);
    async_load_b128(lds_off32(&ldsT[r1f * DV + c4]), &tb[r1f * DV + c4]);
  }
  WAIT_ASYNC_LE(0x0);
  __syncthreads();

  v8f acc = {};
  // A = Q^T tile [M=n 16][K=kdim 4], B = ktv tile [K=kdim 4][N=vdim 16]
#pragma unroll
  for (int kk = 0; kk < DK; kk += 4) {
    const int r0 = kk + half * 2;
    v2f a, bm;
    a.x  = ldsQ[(r0 + 0) * 64 + ni * 16 + l];
    a.y  = ldsQ[(r0 + 1) * 64 + ni * 16 + l];
    bm.x = ldsT[(r0 + 0) * DV + vi * 16 + l];
    bm.y = ldsT[(r0 + 1) * DV + vi * 16 + l];
    acc = __builtin_amdgcn_wmma_f32_16x16x4_f32(
        false, a, false, bm, (short)0, acc, false, false);
  }

  float* ob = out + (size_t)b * NSEQ * DV;
#pragma unroll
  for (int j = 0; j < 8; ++j)
    ob[(size_t)(nbase + ni * 16 + j + 8 * half) * DV + vi * 16 + l] = acc[j];
}

// ---------------------------------------------------------------------------
extern "C" void kernel_launch(void* const* d_in, const int* in_sizes, int n_in,
                              void* d_out, int out_size, void* d_ws,
                              size_t ws_size, hipStream_t stream) {
  const float* q = (const float*)d_in[0];  // [B, DK, N]
  const float* k = (const float*)d_in[1];  // [B, N, DK]
  const float* v = (const float*)d_in[2];  // [B, N, DV]
  float* out = (float*)d_out;              // [B, N, DV]

  float* ws      = (float*)d_ws;
  float* partial = ws;                                   // B*SPLITS*64*64 = 4 MB
  float* ktv     = ws + (size_t)NB * SPLITS * DK * DV;   // B*64*64 = 256 KB

  ktv_partial_kernel<<<NB * SPLITS, 512, 0, stream>>>(k, v, partial);
  ktv_reduce_kernel<<<NB, 256, 0, stream>>>(partial, ktv);
  qktv_kernel<<<dim3(NSEQ / 64, NB), 512, 0, stream>>>(q, ktv, out);
}